// PhysicsDeepwave_17480516894853
// MI455X (gfx1250) — compile-verified
//
#include <hip/hip_runtime.h>

typedef float v2f __attribute__((ext_vector_type(2)));
typedef float v8f __attribute__((ext_vector_type(8)));

namespace dw {
constexpr int NY0 = 100, NX0 = 200;      // unpadded model
constexpr int PML = 35;
constexpr int NYP = NY0 + 2 * PML;       // 170 padded rows
constexpr int NXP = NX0 + 2 * PML;       // 270 padded cols
constexpr int NSH = 2;                   // shots
constexpr int NT  = 300;
constexpr int NREC = 100;
constexpr int G   = NYP * NXP;           // 45900 cells per shot
constexpr int F   = NSH * G;             // 91800 per field (all shots)
constexpr int TY  = (NYP + 15) / 16;     // 11
constexpr int TX  = (NXP + 15) / 16;     // 17
constexpr int NTILE = NSH * TY * TX;     // 374 tile-waves
constexpr int NB = (NTILE + 3) / 4;      // 4 tile-waves per 128-thread block
constexpr float DT  = 1e-3f;
constexpr float DT2 = 1e-6f;

// Banded coefficient fragments precomputed at COMPILE TIME into device rodata.
// Layout matches WMMA f32 16x16x4 A/B fragment striping:
//   lane holds W[k0 + {0,1} - rc], rc = lane&15, k0 = 4*kk + (lane&16 ? 2 : 0).
struct alignas(16) FTab { float v[6][32][2]; };
constexpr double C2R[9] = {-1.0/560.0, 8.0/315.0, -0.2, 1.6, -205.0/72.0,
                           1.6, -0.2, 8.0/315.0, -1.0/560.0};
constexpr double C1R[9] = { 1.0/280.0, -4.0/105.0, 0.2, -0.8, 0.0,
                            0.8, -0.2, 4.0/105.0, -1.0/280.0};
constexpr FTab mk(const double* c, double scale) {
  FTab t{};
  for (int kk = 0; kk < 6; ++kk)
    for (int lane = 0; lane < 32; ++lane) {
      int rc = lane & 15;
      int k0 = 4 * kk + ((lane & 16) ? 2 : 0);
      for (int e = 0; e < 2; ++e) {
        int d = k0 + e - rc;
        t.v[kk][lane][e] = (d >= 0 && d < 9) ? (float)(c[d] * scale) : 0.0f;
      }
    }
  return t;
}
} // namespace dw

__device__ const dw::FTab g_f2 = dw::mk(dw::C2R, 0.01);  // C2 / DX^2
__device__ const dw::FTab g_f1 = dw::mk(dw::C1R, 0.1);   // C1 / DX

__device__ __forceinline__ v2f frag_of(const dw::FTab& t, int kk, int lane) {
  return *(const v2f*)&t.v[kk][lane][0];   // 8B-aligned -> global_load_b64
}

__device__ __forceinline__ v8f wmma4(v2f a, v2f b, v8f c) {
  return __builtin_amdgcn_wmma_f32_16x16x4_f32(false, a, false, b, (short)0, c, false, false);
}

// Zero-padded (exterior) or direct (interior) load from a [NYP x NXP] field.
template <bool IN>
__device__ __forceinline__ float ldf(const float* __restrict__ g, int y, int x) {
  if (IN) return g[y * dw::NXP + x];
  bool ok = ((unsigned)y < (unsigned)dw::NYP) && ((unsigned)x < (unsigned)dw::NXP);
  int idx = (ok ? y : 0) * dw::NXP + (ok ? x : 0);
  float v = g[idx];
  return ok ? v : 0.0f;
}

// A-operand (16x4) for x-direction stencil: rows = tile rows, K = patch columns.
template <bool IN>
__device__ __forceinline__ v2f load_ax(const float* __restrict__ g, int y0, int x0,
                                       int rc, int k0) {
  int y = y0 + rc, xb = x0 - 4 + k0;
  v2f r; r.x = ldf<IN>(g, y, xb); r.y = ldf<IN>(g, y, xb + 1); return r;
}
// B-operand (4x16) for y-direction stencil: K = patch rows, cols = tile cols.
template <bool IN>
__device__ __forceinline__ v2f load_by(const float* __restrict__ g, int y0, int x0,
                                       int rc, int k0) {
  int x = x0 + rc, yb = y0 - 4 + k0;
  v2f r; r.x = ldf<IN>(g, yb, x); r.y = ldf<IN>(g, yb + 1, x); return r;
}

__device__ __forceinline__ bool tile_setup(int& s, int& y0, int& x0) {
  int wid = blockIdx.x * (blockDim.x >> 5) + (threadIdx.x >> 5);
  if (wid >= dw::NTILE) return false;
  s = wid / (dw::TY * dw::TX);
  int r = wid % (dw::TY * dw::TX);
  y0 = (r / dw::TX) * 16;
  x0 = (r % dw::TX) * 16;
  return true;
}

// ---- Phase A tile body: py' = by*py + (by-1)*D1y(u),  px' analogous ----
template <bool IN>
__device__ __forceinline__ void phaseA_tile(const float* __restrict__ us,
                                            float* __restrict__ pys, float* __restrict__ pxs,
                                            const float* __restrict__ by,
                                            const float* __restrict__ bx,
                                            int y0, int x0, int lane) {
  using namespace dw;
  int rc = lane & 15;
  int kb = (lane & 16) ? 2 : 0;
  v8f dy = {0,0,0,0,0,0,0,0};
  v8f dx = {0,0,0,0,0,0,0,0};
#pragma unroll
  for (int kk = 0; kk < 6; ++kk) {
    int k0 = 4 * kk + kb;
    v2f w1 = frag_of(g_f1, kk, lane);
    dy = wmma4(w1, load_by<IN>(us, y0, x0, rc, k0), dy);   // Wy1(16x24) * U(24x16)
    dx = wmma4(load_ax<IN>(us, y0, x0, rc, k0), w1, dx);   // U(16x24) * Wx1(24x16)
  }
  int x = x0 + rc;
  int mb = (lane & 16) ? 8 : 0;
#pragma unroll
  for (int vi = 0; vi < 8; ++vi) {
    int y = y0 + mb + vi;
    bool ok = IN || ((y < NYP) && (x < NXP));
    int idx = (ok ? y : 0) * NXP + (ok ? x : 0);
    float byv = by[ok ? y : 0];
    float bxv = bx[ok ? x : 0];
    float pyn = byv * pys[idx] + (byv - 1.0f) * dy[vi];
    float pxn = bxv * pxs[idx] + (bxv - 1.0f) * dx[vi];
    if (ok) { pys[idx] = pyn; pxs[idx] = pxn; }
  }
}

// ---- Phase B tile body: lap = D2y(u)+D2x(u)+D1y(py)+D1x(px); leapfrog into um ----
template <bool IN>
__device__ __forceinline__ void phaseB_tile(const float* __restrict__ us, float* __restrict__ ums,
                                            const float* __restrict__ pys,
                                            const float* __restrict__ pxs,
                                            const float* __restrict__ v2,
                                            const float* __restrict__ sy,
                                            const float* __restrict__ sx,
                                            int sly, int slx, float sval,
                                            int y0, int x0, int lane) {
  using namespace dw;
  int rc = lane & 15;
  int kb = (lane & 16) ? 2 : 0;
  v8f acc = {0,0,0,0,0,0,0,0};           // all four stencil products chain into one C
#pragma unroll
  for (int kk = 0; kk < 6; ++kk) {
    int k0 = 4 * kk + kb;
    v2f w2 = frag_of(g_f2, kk, lane);
    v2f w1 = frag_of(g_f1, kk, lane);
    acc = wmma4(load_ax<IN>(us,  y0, x0, rc, k0), w2, acc);   // D2x(u)
    acc = wmma4(w2, load_by<IN>(us,  y0, x0, rc, k0), acc);   // D2y(u)
    acc = wmma4(load_ax<IN>(pxs, y0, x0, rc, k0), w1, acc);   // D1x(px)
    acc = wmma4(w1, load_by<IN>(pys, y0, x0, rc, k0), acc);   // D1y(py)
  }
  int x = x0 + rc;
  int mb = (lane & 16) ? 8 : 0;
#pragma unroll
  for (int vi = 0; vi < 8; ++vi) {
    int y = y0 + mb + vi;
    bool ok = IN || ((y < NYP) && (x < NXP));
    int idx = (ok ? y : 0) * NXP + (ok ? x : 0);
    float uv  = us[idx];
    float umv = ums[idx];
    float v2v = v2[idx];
    float syv = sy[ok ? y : 0];
    float sxv = sx[ok ? x : 0];
    float sv  = syv + sxv;
    float prv = syv * sxv;
    float f   = ((y == sly) && (x == slx)) ? sval : 0.0f;
    float un  = ((2.0f - prv * DT2) * uv - (1.0f - 0.5f * sv * DT) * umv
                 + DT2 * v2v * (acc[vi] + f)) / (1.0f + 0.5f * sv * DT);
    if (ok) ums[idx] = un;               // un overwrites old um (pointwise-only dep)
  }
}

// ---- Phase A kernel; last block records previous step's receivers from u==un(t-1) ----
__global__ void dw_phaseA(const float* __restrict__ u, float* __restrict__ py,
                          float* __restrict__ px, const float* __restrict__ by,
                          const float* __restrict__ bx, const int* __restrict__ recloc,
                          float* __restrict__ out, int tprev) {
  using namespace dw;
  if (blockIdx.x == (unsigned)NB) {      // folded receiver recording
    if (tprev >= 0) {
      for (int r = threadIdx.x; r < NSH * NREC; r += blockDim.x) {
        int s = r / NREC, j = r % NREC;
        int ry = recloc[(s * NREC + j) * 2 + 0] + PML;
        int rx = recloc[(s * NREC + j) * 2 + 1] + PML;
        out[(s * NT + tprev) * NREC + j] = u[s * G + ry * NXP + rx];
      }
    }
    return;
  }
  int s, y0, x0;
  if (!tile_setup(s, y0, x0)) return;
  int lane = threadIdx.x & 31;
  bool in = (y0 > 0) && (y0 + 20 <= NYP) && (x0 > 0) && (x0 + 20 <= NXP);
  if (in) phaseA_tile<true >(u + s*G, py + s*G, px + s*G, by, bx, y0, x0, lane);
  else    phaseA_tile<false>(u + s*G, py + s*G, px + s*G, by, bx, y0, x0, lane);
}

__global__ void dw_phaseB(const float* __restrict__ u, float* __restrict__ um,
                          const float* __restrict__ py, const float* __restrict__ px,
                          const float* __restrict__ v2, const float* __restrict__ sy,
                          const float* __restrict__ sx, const float* __restrict__ src,
                          const int* __restrict__ srcloc, int t) {
  using namespace dw;
  int s, y0, x0;
  if (!tile_setup(s, y0, x0)) return;
  int lane = threadIdx.x & 31;
  int sly = srcloc[s * 2 + 0] + PML;     // n_src == 1
  int slx = srcloc[s * 2 + 1] + PML;
  float sval = src[s * NT + t];
  bool in = (y0 > 0) && (y0 + 20 <= NYP) && (x0 > 0) && (x0 + 20 <= NXP);
  if (in) phaseB_tile<true >(u + s*G, um + s*G, py + s*G, px + s*G, v2, sy, sx,
                             sly, slx, sval, y0, x0, lane);
  else    phaseB_tile<false>(u + s*G, um + s*G, py + s*G, px + s*G, v2, sy, sx,
                             sly, slx, sval, y0, x0, lane);
}

// ---- Standalone record for the final step ----
__global__ void dw_record(const float* __restrict__ un, const int* __restrict__ recloc,
                          float* __restrict__ out, int t) {
  using namespace dw;
  int r = blockIdx.x * blockDim.x + threadIdx.x;
  if (r >= NSH * NREC) return;
  int s = r / NREC, j = r % NREC;
  int ry = recloc[(s * NREC + j) * 2 + 0] + PML;
  int rx = recloc[(s * NREC + j) * 2 + 1] + PML;
  out[(s * NT + t) * NREC + j] = un[s * G + ry * NXP + rx];
}

// ---- Init: zero state (u,um,py,px contiguous) and build v2 = edge-padded vp^2 ----
__global__ void dw_init(float* __restrict__ state4, const float* __restrict__ vp,
                        float* __restrict__ v2) {
  using namespace dw;
  int i = blockIdx.x * blockDim.x + threadIdx.x;
  int stride = gridDim.x * blockDim.x;
  for (int k = i; k < 4 * F; k += stride) state4[k] = 0.0f;
  for (int k = i; k < G; k += stride) {
    int y = k / NXP, x = k % NXP;
    int vy = min(max(y - PML, 0), NY0 - 1);
    int vx = min(max(x - PML, 0), NX0 - 1);
    float v = vp[vy * NX0 + vx];
    v2[k] = v * v;
  }
}

// ---- Setup: vmax reduction + quadratic CPML profiles and recursion factors ----
__global__ void dw_setup(const float* __restrict__ vp, float* __restrict__ sy,
                         float* __restrict__ sx, float* __restrict__ by,
                         float* __restrict__ bx) {
  using namespace dw;
  __shared__ float red[256];
  int tid = threadIdx.x;
  float m = 0.0f;
  for (int i = tid; i < NY0 * NX0; i += 256) m = fmaxf(m, vp[i]);
  red[tid] = m;
  __syncthreads();
  for (int off = 128; off > 0; off >>= 1) {
    if (tid < off) red[tid] = fmaxf(red[tid], red[tid + off]);
    __syncthreads();
  }
  float vmax = red[0];
  float smax = 3.0f * vmax * 6.907755279f / (2.0f * (float)PML * 10.0f);  // ln(1000)
  for (int i = tid; i < NYP; i += 256) {
    float d = (float)i;
    float r = fmaxf(fmaxf(((float)PML - d) / (float)PML,
                          (d - (float)(NYP - 1 - PML)) / (float)PML), 0.0f);
    float sg = smax * r * r;
    sy[i] = sg;
    by[i] = expf(-sg * DT);
  }
  for (int i = tid; i < NXP; i += 256) {
    float d = (float)i;
    float r = fmaxf(fmaxf(((float)PML - d) / (float)PML,
                          (d - (float)(NXP - 1 - PML)) / (float)PML), 0.0f);
    float sg = smax * r * r;
    sx[i] = sg;
    bx[i] = expf(-sg * DT);
  }
}

extern "C" void kernel_launch(void* const* d_in, const int* in_sizes, int n_in,
                              void* d_out, int out_size, void* d_ws, size_t ws_size,
                              hipStream_t stream) {
  using namespace dw;
  (void)in_sizes; (void)n_in; (void)out_size; (void)ws_size;
  const float* vp     = (const float*)d_in[0];   // [100,200]
  const float* src    = (const float*)d_in[1];   // [2,1,300]
  const int*   srcloc = (const int*)d_in[2];     // [2,1,2]
  const int*   recloc = (const int*)d_in[3];     // [2,100,2]
  float* out = (float*)d_out;                    // [1,2,300,100]
  float* ws  = (float*)d_ws;

  float* bufA = ws;            // u   (t even)
  float* bufB = ws + F;        // um  (t even)
  float* py   = ws + 2 * F;
  float* px   = ws + 3 * F;
  float* v2   = ws + 4 * F;
  float* sy   = v2 + G;
  float* sx   = sy + NYP;
  float* by   = sx + NXP;
  float* bx   = by + NYP;

  dw_init<<<512, 256, 0, stream>>>(ws, vp, v2);
  dw_setup<<<1, 256, 0, stream>>>(vp, sy, sx, by, bx);

  for (int t = 0; t < NT; ++t) {
    const float* u = (t & 1) ? bufB : bufA;
    float* um      = (t & 1) ? bufA : bufB;
    // phaseA's extra block records step t-1's receivers (u == un(t-1))
    dw_phaseA<<<NB + 1, 128, 0, stream>>>(u, py, px, by, bx, recloc, out, t - 1);
    dw_phaseB<<<NB, 128, 0, stream>>>(u, um, py, px, v2, sy, sx, src, srcloc, t);
  }
  float* unLast = ((NT - 1) & 1) ? bufA : bufB;  // un of final step
  dw_record<<<1, 256, 0, stream>>>(unLast, recloc, out, NT - 1);
}